// CosineSSMLoss_25426206392656
// MI455X (gfx1250) — compile-verified
//
#include <hip/hip_runtime.h>
#include <hip/hip_bf16.h>

typedef __attribute__((ext_vector_type(2))) float v2f;
typedef __attribute__((ext_vector_type(8))) float v8f;

#define N_SPATIAL 4096   // H*W = 64*64
#define C_CH      4
#define INV_EPS   1e12f  // 1 / F.normalize eps (1e-12)

// One block per batch pair {2p, 2p+1}. Builds Y (16 x N): rows 0-3 pred[b0],
// 4-7 src[b0], 8-11 pred[b1], 12-15 src[b1] (columns L2-normalized over C),
// accumulates G = Y*Y^T with v_wmma_f32_16x16x4_f32, then reduces the signed
// squared Frobenius contributions to a single float per block.
__global__ __launch_bounds__(256) void cosine_gram_loss_kernel(
    const float* __restrict__ xp, const float* __restrict__ xs,
    float* __restrict__ ws_out) {
  const int pair = blockIdx.x;        // 0 or 1
  const int tid  = threadIdx.x;       // 0..255
  const int wave = tid >> 5;          // 0..7
  const int lane = tid & 31;

  // WMMA operand layout for 16x16x4 f32 Gram (A and B per-lane data coincide):
  // lane L supplies Y[m, k0 + koff + v], m = L&15, koff = (L>=16)?2:0.
  const int m    = lane & 15;
  const int koff = (lane >> 4) << 1;  // 0 or 2
  const int b    = pair * 2 + (m >> 3);
  const int c    = m & 3;
  const float* __restrict__ X = ((m >> 2) & 1) ? xs : xp;
  const float* __restrict__ col = X + (size_t)(b * C_CH) * N_SPATIAL;

  v8f acc = {};  // 16x16 f32 accumulator (8 VGPRs)

  const int k_begin = wave * (N_SPATIAL / 8);
  const int k_end   = k_begin + (N_SPATIAL / 8);
#pragma unroll 4
  for (int k0 = k_begin; k0 < k_end; k0 += 4) {
    const int k = k0 + koff;
    // Load all 4 channels of columns k, k+1 (B64 loads) to build the norm.
    float2 r0 = *(const float2*)(col + 0 * N_SPATIAL + k);
    float2 r1 = *(const float2*)(col + 1 * N_SPATIAL + k);
    float2 r2 = *(const float2*)(col + 2 * N_SPATIAL + k);
    float2 r3 = *(const float2*)(col + 3 * N_SPATIAL + k);

    float ss0 = r0.x * r0.x + r1.x * r1.x + r2.x * r2.x + r3.x * r3.x;
    float ss1 = r0.y * r0.y + r1.y * r1.y + r2.y * r2.y + r3.y * r3.y;
    // 1/max(sqrt(ss), eps) == min(rsqrt(ss), 1/eps)  (ss >= 0; ss==0 -> 1e12)
    float inv0 = fminf(__builtin_amdgcn_rsqf(ss0), INV_EPS);
    float inv1 = fminf(__builtin_amdgcn_rsqf(ss1), INV_EPS);

    float mx = (c == 0) ? r0.x : (c == 1) ? r1.x : (c == 2) ? r2.x : r3.x;
    float my = (c == 0) ? r0.y : (c == 1) ? r1.y : (c == 2) ? r2.y : r3.y;

    v2f a;
    a[0] = mx * inv0;
    a[1] = my * inv1;

    // G += A * A^T   (A operand == B operand for symmetric Gram layout)
    acc = __builtin_amdgcn_wmma_f32_16x16x4_f32(
        /*neg_a=*/false, a, /*neg_b=*/false, a,
        /*c_mod=*/(short)0, acc, /*reuse_a=*/false, /*reuse_b=*/false);
  }

  // ---- cross-wave reduction of the 16x16 partial Grams ----
  __shared__ float lds[8 * 256];
#pragma unroll
  for (int r = 0; r < 8; ++r) lds[wave * 256 + r * 32 + lane] = acc[r];
  __syncthreads();

  // Thread `tid` owns Gram entry e = tid: r = tid>>5, lane' = tid&31.
  float d = 0.0f;
#pragma unroll
  for (int w = 0; w < 8; ++w) d += lds[w * 256 + tid];

  const int r  = tid >> 5;
  const int lp = tid & 31;
  const int gi = r + ((lp >> 4) ? 8 : 0);  // Gram row
  const int gj = lp & 15;                  // Gram col
  // Same batch pair block?  (rows/cols 0-7 = b0, 8-15 = b1)
  const bool same_batch = ((gi >> 3) == (gj >> 3));
  // pred-pred / src-src -> +1 ; pred-src cross -> -1
  const float sgn = (((gi >> 2) & 1) == ((gj >> 2) & 1)) ? 1.0f : -1.0f;
  float contrib = same_batch ? sgn * d * d : 0.0f;

  // wave32 shuffle reduction, then LDS across the 8 waves
#pragma unroll
  for (int off = 16; off > 0; off >>= 1)
    contrib += __shfl_down(contrib, off, 32);
  __syncthreads();
  if (lp == 0) lds[tid >> 5] = contrib;
  __syncthreads();
  if (tid == 0) {
    float s = 0.0f;
#pragma unroll
    for (int w = 0; w < 8; ++w) s += lds[w];
    ws_out[blockIdx.x] = s;
  }
}

__global__ void cosine_gram_loss_finalize(const float* __restrict__ ws,
                                          float* __restrict__ out) {
  // loss = (pair0 + pair1) / (B * N * N) = sum / 67108864
  out[0] = (ws[0] + ws[1]) * (1.0f / 67108864.0f);
}

extern "C" void kernel_launch(void* const* d_in, const int* in_sizes, int n_in,
                              void* d_out, int out_size, void* d_ws, size_t ws_size,
                              hipStream_t stream) {
  const float* xp = (const float*)d_in[0];
  const float* xs = (const float*)d_in[1];
  float* out = (float*)d_out;
  float* ws  = (float*)d_ws;

  cosine_gram_loss_kernel<<<dim3(2), dim3(256), 0, stream>>>(xp, xs, ws);
  cosine_gram_loss_finalize<<<dim3(1), dim3(1), 0, stream>>>(ws, out);
}